// Hash_grid_81793357185740
// MI455X (gfx1250) — compile-verified
//
#include <hip/hip_runtime.h>
#include <math.h>
#include <stdint.h>

// ---- Hash-grid static config (mirrors reference) ----
#define NUM_LEVELS   16
#define LEVEL_DIM    2
#define BASE_RES     16
#define LOG2_HASHMAP 19
#define DESIRED_RES  2048
#define MAX_LEVELS   12
#define BLOCK        256

#define PRIME1 2654435761u
#define PRIME2 805459861u
#define HASH_MASK ((1u << LOG2_HASHMAP) - 1u)   // all hashed levels have hsize == 2^19

typedef __attribute__((ext_vector_type(4))) float v4f;   // native vector for NT b128 stores

// Static structure of this config: level is dense iff res^3 <= 2^19 (res <= 80).
// res(l) = ceil(16*2^(7l/15) - 1) + 1 -> 16,23,31,43,59 dense; l>=5 (res>=81) hashed.
__device__ __constant__ constexpr bool kDense[NUM_LEVELS] =
    { true, true, true, true, true,
      false, false, false, false, false, false, false, false, false, false, false };

struct HGParams {
    unsigned offset[NUM_LEVELS];
    unsigned res[NUM_LEVELS];
    unsigned res2[NUM_LEVELS];    // res*res (dense indexing)
    float    scale[NUM_LEVELS];
};

#define L0_ENTRIES 4096   // level 0: dense 16^3 grid, 8-aligned -> 32 KB of float2

__device__ __forceinline__ unsigned uminu(unsigned a, unsigned b) { return a < b ? a : b; }

__global__ __launch_bounds__(BLOCK) void hashgrid_fwd(
    const float*  __restrict__ x,
    const float2* __restrict__ table,
    const int*    __restrict__ step_ptr,
    float*        __restrict__ out,
    int n, HGParams p)
{
    __shared__ float2 s_l0[L0_ENTRIES];   // 32 KB: level-0 table staged in LDS

    const int tid = threadIdx.x;

    // ---- Stage level-0 table into LDS with gfx1250 async global->LDS DMA ----
    // 4096 float2 = 32 KB; 256 threads x 8 iters x 16 B (b128).
    {
        const float2* t0 = table + p.offset[0];
        #pragma unroll
        for (int it = 0; it < 8; ++it) {
            int e = (it * BLOCK + tid) * 2;                     // float2 pair index
            uint32_t lds_off = (uint32_t)(uintptr_t)(&s_l0[e]); // low 32 bits of flat LDS ptr = LDS offset
            uint64_t gaddr   = (uint64_t)(uintptr_t)(t0 + e);
            asm volatile("global_load_async_to_lds_b128 %0, %1, off"
                         :: "v"(lds_off), "v"(gaddr) : "memory");
        }
        asm volatile("s_wait_asynccnt 0" ::: "memory");
    }
    __syncthreads();

    const int i = blockIdx.x * BLOCK + tid;
    if (i >= n) return;

    // step is a device scalar; uniform scalar load + uniform branches per level
    const int step = *step_ptr;
    int active = step / 1000 + 1;
    if (active > MAX_LEVELS) active = MAX_LEVELS;

    // Load point (streamed once -> non-temporal), map [-1,1] -> [0,1] with clamp
    const float* xi = x + 3 * (size_t)i;
    float rx = __builtin_nontemporal_load(xi + 0);
    float ry = __builtin_nontemporal_load(xi + 1);
    float rz = __builtin_nontemporal_load(xi + 2);
    float ux = fminf(fmaxf((rx + 1.0f) * 0.5f, 0.0f), 1.0f);
    float uy = fminf(fmaxf((ry + 1.0f) * 0.5f, 0.0f), 1.0f);
    float uz = fminf(fmaxf((rz + 1.0f) * 0.5f, 0.0f), 1.0f);

    float feat[2 * NUM_LEVELS];
    #pragma unroll
    for (int k = 0; k < 2 * NUM_LEVELS; ++k) feat[k] = 0.0f;

    #pragma unroll
    for (int l = 0; l < NUM_LEVELS; ++l) {
        if (l >= active) continue;                  // uniform (SGPR) branch

        const float    s    = p.scale[l];
        const unsigned res  = p.res[l];
        const unsigned res2 = p.res2[l];
        const unsigned base = p.offset[l];
        const unsigned rmax = res - 1u;

        // pos = u*scale + 0.5 (align_corners=False); pos >= 0.5 so floor >= 0
        float pxf = fmaf(ux, s, 0.5f);
        float pyf = fmaf(uy, s, 0.5f);
        float pzf = fmaf(uz, s, 0.5f);
        float flx = floorf(pxf), fly = floorf(pyf), flz = floorf(pzf);
        float fx = pxf - flx, fy = pyf - fly, fz = pzf - flz;
        float gx = 1.0f - fx, gy = 1.0f - fy, gz = 1.0f - fz;
        unsigned ix = (unsigned)(int)flx;
        unsigned iy = (unsigned)(int)fly;
        unsigned iz = (unsigned)(int)flz;

        // Two clamped coords per axis, pre-scaled: 4 muls/level instead of 16
        unsigned x0 = uminu(ix, rmax),      x1 = uminu(ix + 1u, rmax);
        unsigned y0 = uminu(iy, rmax),      y1 = uminu(iy + 1u, rmax);
        unsigned z0 = uminu(iz, rmax),      z1 = uminu(iz + 1u, rmax);
        unsigned hy0, hy1, hz0, hz1;
        if (kDense[l]) {               // compile-time (unrolled loop)
            hy0 = y0 * res;  hy1 = y1 * res;
            hz0 = z0 * res2; hz1 = z1 * res2;
        } else {
            hy0 = y0 * PRIME1; hy1 = y1 * PRIME1;
            hz0 = z0 * PRIME2; hz1 = z1 * PRIME2;
        }

        // 4 shared xy weight partials, then 1 mul per corner
        float wxy00 = gx * gy, wxy01 = gx * fy, wxy10 = fx * gy, wxy11 = fx * fy;

        float f0 = 0.0f, f1 = 0.0f;
        #pragma unroll
        for (int c = 0; c < 8; ++c) {
            // CORNERS order: (i,j,k) = (c>>2, (c>>1)&1, c&1) on dims (x,y,z)
            unsigned hx = (c & 4) ? x1 : x0;
            unsigned hy = (c & 2) ? hy1 : hy0;
            unsigned hz = (c & 1) ? hz1 : hz0;
            float wxy = (c & 4) ? ((c & 2) ? wxy11 : wxy10)
                                : ((c & 2) ? wxy01 : wxy00);
            float w = wxy * ((c & 1) ? fz : gz);

            unsigned idx = kDense[l] ? (hx + hy + hz)
                                     : ((hx ^ hy ^ hz) & HASH_MASK);

            float2 v;
            if (l == 0) {
                v = s_l0[idx];                       // ds_load_b64 from staged LDS
            } else {
                v = table[base + idx];               // global_load_b64 gather (L2-resident)
            }
            f0 = fmaf(w, v.x, f0);
            f1 = fmaf(w, v.y, f1);
        }
        feat[2 * l]     = f0;
        feat[2 * l + 1] = f1;
    }

    // 128-bit NON-TEMPORAL stores: write-once output must not evict the
    // L2-resident gather table (51 MB table vs 256 MB output stream).
    v4f* o4 = (v4f*)(out + (size_t)i * (2 * NUM_LEVELS));
    #pragma unroll
    for (int k = 0; k < 8; ++k) {
        v4f v4 = { feat[4 * k], feat[4 * k + 1], feat[4 * k + 2], feat[4 * k + 3] };
        __builtin_nontemporal_store(v4, &o4[k]);
    }
}

extern "C" void kernel_launch(void* const* d_in, const int* in_sizes, int n_in,
                              void* d_out, int out_size, void* d_ws, size_t ws_size,
                              hipStream_t stream) {
    const float*  x     = (const float*)d_in[0];
    const float2* table = (const float2*)d_in[1];
    const int*    step  = (const int*)d_in[2];
    float*        out   = (float*)d_out;
    const int n = in_sizes[0] / 3;

    // Host-side replication of level_meta() in double precision
    HGParams p;
    const double b = exp2(log2((double)DESIRED_RES / (double)BASE_RES) /
                          (double)(NUM_LEVELS - 1));
    unsigned long long off = 0ull;
    const unsigned long long hmax = 1ull << LOG2_HASHMAP;
    for (int l = 0; l < NUM_LEVELS; ++l) {
        double scale = pow(b, (double)l) * (double)BASE_RES - 1.0;
        unsigned res = (unsigned)ceil(scale) + 1u;
        unsigned long long n3 = (unsigned long long)res * res * res;
        unsigned long long ne = n3 < hmax ? n3 : hmax;
        ne = ((ne + 7ull) / 8ull) * 8ull;          // align to 8
        p.offset[l] = (unsigned)off;
        p.res[l]    = res;
        p.res2[l]   = res * res;
        p.scale[l]  = (float)scale;
        off += ne;
    }

    dim3 grid((n + BLOCK - 1) / BLOCK), block(BLOCK);
    hipLaunchKernelGGL(hashgrid_fwd, grid, block, 0, stream,
                       x, table, step, out, n, p);
}